// AnchorPositionExtractor_24687472017806
// MI455X (gfx1250) — compile-verified
//
#include <hip/hip_runtime.h>
#include <hip/hip_bf16.h>
#include <float.h>
#include <math.h>

// ---------------------------------------------------------------------------
// AnchorPositionExtractor — CDNA5 / gfx1250 (wave32, WMMA f32 16x16x4)
// ---------------------------------------------------------------------------

constexpr int B  = 32;
constexpr int N  = 8192;
constexpr int E  = 256;
constexpr int NA = 16;
constexpr float PAD_TOKEN = -2.0f;
constexpr float LN_EPS    = 1e-3f;

constexpr int ROWS_PER_WAVE = 128;
constexpr int CHUNKS = N / ROWS_PER_WAVE;   // 64 per batch

// workspace layout (float offsets, all 256B-aligned)
constexpr size_t OFF_QK     = 0;
constexpr size_t OFF_LOGITS = 256;
constexpr size_t OFF_ATT    = OFF_LOGITS + (size_t)B * N;
constexpr size_t OFF_PM     = OFF_ATT + (size_t)B * N;
constexpr size_t OFF_PS     = OFF_PM + (size_t)B * CHUNKS;
constexpr size_t OFF_PV     = OFF_PS + (size_t)B * CHUNKS;
constexpr size_t OFF_GM     = OFF_PV + (size_t)B * CHUNKS * E;
constexpr size_t OFF_DEN    = OFF_GM + 64;
constexpr size_t OFF_S      = OFF_DEN + 64;
constexpr size_t OFF_INDS   = OFF_S + (size_t)B * E;      // int32 region
constexpr size_t OFF_OUTS0  = OFF_INDS + (size_t)B * NA;
constexpr size_t OFF_PRELN  = OFF_OUTS0 + (size_t)B * NA * E;

typedef __attribute__((ext_vector_type(2))) float v2f;
typedef __attribute__((ext_vector_type(8))) float v8f;

__device__ __forceinline__ v8f wmma4(v2f a, v2f b, v8f c) {
    // D = A(16x4,f32) * B(4x16,f32) + C(16x16,f32)
    return __builtin_amdgcn_wmma_f32_16x16x4_f32(
        /*neg_a=*/false, a, /*neg_b=*/false, b,
        /*c_mod=*/(short)0, c, /*reuse_a=*/false, /*reuse_b=*/false);
}

// K0: q = barcode @ Wq ; qk = scale * (Wk @ q)
__global__ void k_qk(const float* __restrict__ bc, const float* __restrict__ Wq,
                     const float* __restrict__ Wk, float* __restrict__ ws) {
    __shared__ float qs[E];
    int t = threadIdx.x;
    float acc = 0.f;
    for (int f = 0; f < E; ++f) acc += bc[f] * Wq[f * E + t];
    qs[t] = acc;
    __syncthreads();
    float a2 = 0.f;
    for (int e = 0; e < E; ++e) a2 += Wk[t * E + e] * qs[e];
    ws[OFF_QK + t] = a2 * 0.0625f;   // 1/sqrt(256)
}

// K1: single pass over x — logits + online-softmax weighted row accumulation
__global__ void __launch_bounds__(256)
k_flash(const float* __restrict__ x, const float* __restrict__ mask,
        float* __restrict__ ws) {
    int lane = threadIdx.x & 31;
    int wave = threadIdx.x >> 5;
    int gw   = blockIdx.x * 8 + wave;
    int rowBase = gw * ROWS_PER_WAVE;
    int b  = rowBase / N;
    int n0 = rowBase - b * N;
    int chunk = n0 / ROWS_PER_WAVE;

    const float4* qk4 = (const float4*)(ws + OFF_QK);
    float4 q0 = qk4[lane * 2], q1 = qk4[lane * 2 + 1];
    float* logits = ws + OFF_LOGITS;

    float a0=0,a1=0,a2=0,a3=0,a4=0,a5=0,a6=0,a7=0;
    float runM = -FLT_MAX, runS = 0.f;

    for (int r = 0; r < ROWS_PER_WAVE; ++r) {
        int n = n0 + r;
        const float4* xr = (const float4*)(x + ((size_t)b * N + n) * E);
        float4 x0 = xr[lane * 2], x1 = xr[lane * 2 + 1];
        float d = x0.x*q0.x + x0.y*q0.y + x0.z*q0.z + x0.w*q0.w
                + x1.x*q1.x + x1.y*q1.y + x1.z*q1.z + x1.w*q1.w;
        #pragma unroll
        for (int off = 16; off; off >>= 1) d += __shfl_xor(d, off, 32);
        float mv = mask[(size_t)b * N + n];
        float logit = d + ((mv == PAD_TOKEN) ? -1.0e9f : 0.f);
        if (lane == 0) logits[(size_t)b * N + n] = logit;
        float newM = fmaxf(runM, logit);
        float sc = expf(runM - newM);   // first iter: expf(-huge) == 0
        float p  = expf(logit - newM);
        runS = runS * sc + p;
        a0 = a0*sc + p*x0.x; a1 = a1*sc + p*x0.y;
        a2 = a2*sc + p*x0.z; a3 = a3*sc + p*x0.w;
        a4 = a4*sc + p*x1.x; a5 = a5*sc + p*x1.y;
        a6 = a6*sc + p*x1.z; a7 = a7*sc + p*x1.w;
        runM = newM;
    }
    int c = b * CHUNKS + chunk;
    if (lane == 0) { ws[OFF_PM + c] = runM; ws[OFF_PS + c] = runS; }
    float4* pv = (float4*)(ws + OFF_PV + (size_t)c * E);
    pv[lane * 2]     = make_float4(a0, a1, a2, a3);
    pv[lane * 2 + 1] = make_float4(a4, a5, a6, a7);
}

// K2: combine chunk partials -> global max, denom, s[b,:] = sum_n att*x / denom
__global__ void k_combine(float* __restrict__ ws) {
    __shared__ float red[256];
    __shared__ float scl[CHUNKS];
    int t = threadIdx.x, b = blockIdx.x;
    const float* pm = ws + OFF_PM + (size_t)b * CHUNKS;
    const float* ps = ws + OFF_PS + (size_t)b * CHUNKS;

    red[t] = (t < CHUNKS) ? pm[t] : -FLT_MAX;
    __syncthreads();
    for (int s = 128; s; s >>= 1) { if (t < s) red[t] = fmaxf(red[t], red[t+s]); __syncthreads(); }
    float gm = red[0];
    __syncthreads();
    if (t < CHUNKS) scl[t] = expf(pm[t] - gm);
    red[t] = 0.f;
    __syncthreads();
    red[t] = (t < CHUNKS) ? scl[t] * ps[t] : 0.f;
    __syncthreads();
    for (int s = 128; s; s >>= 1) { if (t < s) red[t] += red[t+s]; __syncthreads(); }
    float den = red[0];

    const float* pv = ws + OFF_PV + (size_t)b * CHUNKS * E;
    float acc = 0.f;
    for (int c = 0; c < CHUNKS; ++c) acc += scl[c] * pv[(size_t)c * E + t];
    ws[OFF_S + (size_t)b * E + t] = acc / den;
    if (t == 0) { ws[OFF_GM + b] = gm; ws[OFF_DEN + b] = den; }
}

// K3: materialize att[b,n] = exp(l-gm)*m/den
__global__ void k_att(const float* __restrict__ mask, float* __restrict__ ws) {
    int idx = blockIdx.x * 256 + threadIdx.x;
    int b = idx / N;
    float mv = (mask[idx] == PAD_TOKEN) ? 0.f : 1.f;
    ws[OFF_ATT + idx] = expf(ws[OFF_LOGITS + idx] - ws[OFF_GM + b]) * mv / ws[OFF_DEN + b];
}

// K4: barcode_out = s(32x256) @ Wv(256x256) — WMMA f32 16x16x4, 1 wave / 16-col strip
__global__ void __launch_bounds__(32)
k_barcode(const float* __restrict__ Wv, const float* __restrict__ ws,
          float* __restrict__ out_bar) {
    int lane = threadIdx.x;
    int n0 = blockIdx.x * 16;
    int mrow = lane & 15, khalf = lane >> 4;
    int ncol = n0 + mrow;
    const float* S = ws + OFF_S;
    v8f c0 = {}, c1 = {};
    for (int k0 = 0; k0 < E; k0 += 4) {
        int kb = k0 + khalf * 2;
        float2 A0 = *(const float2*)(S + (size_t)mrow * E + kb);
        float2 A1 = *(const float2*)(S + (size_t)(mrow + 16) * E + kb);
        v2f a0; a0[0] = A0.x; a0[1] = A0.y;
        v2f a1; a1[0] = A1.x; a1[1] = A1.y;
        v2f bb; bb[0] = Wv[(size_t)kb * E + ncol]; bb[1] = Wv[(size_t)(kb + 1) * E + ncol];
        c0 = wmma4(a0, bb, c0);
        c1 = wmma4(a1, bb, c1);
    }
    int f = n0 + (lane & 15);
    int mhi = (lane >> 4) * 8;
    #pragma unroll
    for (int v = 0; v < 8; ++v) {
        int m = v + mhi;
        out_bar[(size_t)m * E + f]        = c0[v];
        out_bar[(size_t)(m + 16) * E + f] = c1[v];
    }
}

// K5: greedy distance-constrained top-NA (iterated argmax == sorted-scan greedy)
__global__ void k_select(float* __restrict__ ws, float* __restrict__ out_inds,
                         float* __restrict__ out_w) {
    __shared__ float attL[N];       // 32 KB
    __shared__ float rv[256];
    __shared__ int   ri[256];
    __shared__ int   sel[NA];
    __shared__ int   nsel;
    int t = threadIdx.x, b = blockIdx.x;
    const float* att = ws + OFF_ATT + (size_t)b * N;
    for (int j = t; j < N; j += 256) attL[j] = att[j];
    if (t < NA) sel[t] = 0;
    if (t == 0) nsel = 0;
    __syncthreads();

    for (int round = 0; round < NA; ++round) {
        int cnt = nsel;
        float bv = -FLT_MAX; int bi = 0;
        for (int j = t; j < N; j += 256) {
            float v = attL[j];
            bool ok = true;
            for (int k = 0; k < cnt; ++k) {
                int d = j - sel[k]; if (d < 0) d = -d;
                if (d < 3) { ok = false; break; }     // M_MIN=3; M_MAX=N always ok
            }
            if (ok && (v > bv || (v == bv && j < bi))) { bv = v; bi = j; }
        }
        rv[t] = bv; ri[t] = bi;
        __syncthreads();
        for (int s = 128; s; s >>= 1) {
            if (t < s) {
                float v2 = rv[t + s]; int i2 = ri[t + s];
                if (v2 > rv[t] || (v2 == rv[t] && i2 < ri[t])) { rv[t] = v2; ri[t] = i2; }
            }
            __syncthreads();
        }
        if (t == 0 && rv[0] > -FLT_MAX && cnt < NA) { sel[cnt] = ri[0]; nsel = cnt + 1; }
        __syncthreads();
    }

    if (t == 0) {   // ascending insertion sort (unfilled slots stay 0)
        for (int i = 1; i < NA; ++i) {
            int key = sel[i], j = i - 1;
            while (j >= 0 && sel[j] > key) { sel[j + 1] = sel[j]; --j; }
            sel[j + 1] = key;
        }
    }
    __syncthreads();
    if (t < NA) {
        int ind = sel[t];
        ((int*)(ws + OFF_INDS))[b * NA + t] = ind;
        out_inds[b * NA + t] = (float)ind;
        out_w[b * NA + t]    = attL[ind];
    }
}

// K6: gather anchors + sinusoidal position
__global__ void k_gather(const float* __restrict__ x, float* __restrict__ ws) {
    int bq = blockIdx.x;          // b*NA + a
    int b  = bq / NA;
    int t  = threadIdx.x;
    int ind = ((const int*)(ws + OFF_INDS))[bq];
    float ra  = (float)t / 256.0f;
    float pos = sinf((float)ind / powf(40.0f, ra));
    ws[OFF_OUTS0 + (size_t)bq * E + t] = x[((size_t)b * N + ind) * E + t] + pos;
}

// K7: per-anchor  outs@w[a]  *  sigmoid(outs@g[a])  * weight  — WMMA f32 16x16x4
__global__ void __launch_bounds__(32)
k_anchor(const float* __restrict__ w, const float* __restrict__ g,
         const float* __restrict__ out_weights, float* __restrict__ ws) {
    int lane = threadIdx.x;
    int n0 = blockIdx.x * 16;
    int a  = blockIdx.y;
    int mrow = lane & 15, khalf = lane >> 4;
    int ncol = n0 + mrow;
    const float* A  = ws + OFF_OUTS0;             // [B][NA][E]
    const float* Wm = w + (size_t)a * E * E;
    const float* Gm = g + (size_t)a * E * E;

    v8f cw0 = {}, cw1 = {}, cg0 = {}, cg1 = {};
    for (int k0 = 0; k0 < E; k0 += 4) {
        int kb = k0 + khalf * 2;
        float2 A0 = *(const float2*)(A + ((size_t)mrow * NA + a) * E + kb);
        float2 A1 = *(const float2*)(A + ((size_t)(mrow + 16) * NA + a) * E + kb);
        v2f a0; a0[0] = A0.x; a0[1] = A0.y;
        v2f a1; a1[0] = A1.x; a1[1] = A1.y;
        v2f bw; bw[0] = Wm[(size_t)kb * E + ncol]; bw[1] = Wm[(size_t)(kb + 1) * E + ncol];
        v2f bg; bg[0] = Gm[(size_t)kb * E + ncol]; bg[1] = Gm[(size_t)(kb + 1) * E + ncol];
        cw0 = wmma4(a0, bw, cw0);
        cw1 = wmma4(a1, bw, cw1);
        cg0 = wmma4(a0, bg, cg0);
        cg1 = wmma4(a1, bg, cg1);
    }
    int f = n0 + (lane & 15);
    int mhi = (lane >> 4) * 8;
    float* preln = ws + OFF_PRELN;
    #pragma unroll
    for (int v = 0; v < 8; ++v) {
        int m = v + mhi;
        float wt0 = out_weights[m * NA + a];
        preln[((size_t)m * NA + a) * E + f] =
            cw0[v] * (1.f / (1.f + expf(-cg0[v]))) * wt0;
        int m2 = m + 16;
        float wt1 = out_weights[m2 * NA + a];
        preln[((size_t)m2 * NA + a) * E + f] =
            cw1[v] * (1.f / (1.f + expf(-cg1[v]))) * wt1;
    }
}

// K8: LayerNorm over E
__global__ void k_ln(const float* __restrict__ ws, const float* __restrict__ gamma,
                     const float* __restrict__ beta, float* __restrict__ out_outs) {
    __shared__ float red[256];
    int bq = blockIdx.x, t = threadIdx.x;
    float v = ws[OFF_PRELN + (size_t)bq * E + t];
    red[t] = v;
    __syncthreads();
    for (int s = 128; s; s >>= 1) { if (t < s) red[t] += red[t + s]; __syncthreads(); }
    float mu = red[0] * (1.f / 256.f);
    __syncthreads();
    float d = v - mu;
    red[t] = d * d;
    __syncthreads();
    for (int s = 128; s; s >>= 1) { if (t < s) red[t] += red[t + s]; __syncthreads(); }
    float var = red[0] * (1.f / 256.f);
    out_outs[(size_t)bq * E + t] = d * rsqrtf(var + LN_EPS) * gamma[t] + beta[t];
}

extern "C" void kernel_launch(void* const* d_in, const int* in_sizes, int n_in,
                              void* d_out, int out_size, void* d_ws, size_t ws_size,
                              hipStream_t stream) {
    const float* x     = (const float*)d_in[0];
    const float* mask  = (const float*)d_in[1];
    const float* bc    = (const float*)d_in[2];
    const float* Wq    = (const float*)d_in[3];
    const float* Wk    = (const float*)d_in[4];
    const float* Wv    = (const float*)d_in[5];
    const float* g     = (const float*)d_in[6];
    const float* w     = (const float*)d_in[7];
    const float* gamma = (const float*)d_in[8];
    const float* beta  = (const float*)d_in[9];
    (void)in_sizes; (void)n_in; (void)out_size; (void)ws_size;

    float* ws  = (float*)d_ws;
    float* out = (float*)d_out;
    float* out_outs = out;                               // (B,NA,E)
    float* out_inds = out + (size_t)B * NA * E;          // (B,NA) as float
    float* out_w    = out_inds + B * NA;                 // (B,NA)
    float* out_bar  = out_w + B * NA;                    // (B,E)

    k_qk<<<1, 256, 0, stream>>>(bc, Wq, Wk, ws);
    k_flash<<<(B * N) / (8 * ROWS_PER_WAVE), 256, 0, stream>>>(x, mask, ws);
    k_combine<<<B, 256, 0, stream>>>(ws);
    k_att<<<(B * N) / 256, 256, 0, stream>>>(mask, ws);
    k_barcode<<<E / 16, 32, 0, stream>>>(Wv, ws, out_bar);
    k_select<<<B, 256, 0, stream>>>(ws, out_inds, out_w);
    k_gather<<<B * NA, 256, 0, stream>>>(x, ws);
    dim3 ga(E / 16, NA);
    k_anchor<<<ga, 32, 0, stream>>>(w, g, out_w, ws);
    k_ln<<<B * NA, 256, 0, stream>>>(ws, gamma, beta, out_outs);
}